// SCNNModel_4294967296233
// MI455X (gfx1250) — compile-verified
//
#include <hip/hip_runtime.h>
#include <hip/hip_bf16.h>
#include <math.h>

// ---------------------------------------------------------------------------
// Spherical U-Net (SCNN) for MI455X / gfx1250, fp32 throughout.
// All matmuls lower to v_wmma_f32_16x16x4_f32 (wave32 WMMA).
// ---------------------------------------------------------------------------

typedef float v2f __attribute__((ext_vector_type(2)));
typedef float v8f __attribute__((ext_vector_type(8)));

#define BATCH   16384
#define NC      45      // spherical-harmonic coefficients
#define NCP     48      // padded (3 WMMA k-steps of 4 per 12 steps)
#define ND      100     // directions
#define NDT     7       // ceil(112/16) N-tiles for GEMM1
#define PI_F    3.14159265358979323846f

__device__ __forceinline__ v8f v8zero() {
    v8f z;
#pragma unroll
    for (int i = 0; i < 8; ++i) z[i] = 0.0f;
    return z;
}

__device__ __forceinline__ v8f wmma4(v2f a, v2f b, v8f c) {
    // 8 args: (neg_a, A, neg_b, B, c_mod, C, reuse_a, reuse_b)
    return __builtin_amdgcn_wmma_f32_16x16x4_f32(false, a, false, b,
                                                 (short)0, c, false, false);
}

// ---------------------------------------------------------------------------
// Prep: pack isft^T / sft^T into exact per-lane B-operand layout.
//   B layout (mirror of A): lanes 0-15 hold N=lane,   v0=K0, v1=K1
//                           lanes16-31 hold N=lane-16, v0=K2, v1=K3
// PB1 [7][12][32][2] : GEMM1 B, K=c (0..47, zero-padded >=45), N=d (0..111,
//                      zero-padded >=100)
// PB2 [3][25][32][2] : GEMM2 B, K=d (0..99 exact), N=c (0..47, zero >=45)
// ---------------------------------------------------------------------------
__global__ void prep_pack(const float* __restrict__ sft,
                          const float* __restrict__ isft,
                          float* __restrict__ pb1,
                          float* __restrict__ pb2) {
    int tid = threadIdx.x + blockIdx.x * blockDim.x;
    int stride = blockDim.x * gridDim.x;
    for (int idx = tid; idx < 7 * 12 * 32 * 2; idx += stride) {
        int j    = idx & 1;
        int lane = (idx >> 1) & 31;
        int ks   = (idx >> 6) % 12;
        int nt   = idx / (12 * 64);
        int c    = ks * 4 + ((lane >> 4) << 1) + j;   // K index (coeff)
        int d    = nt * 16 + (lane & 15);             // N index (direction)
        pb1[idx] = (c < NC && d < ND) ? isft[d * NC + c] : 0.0f;
    }
    for (int idx = tid; idx < 3 * 25 * 32 * 2; idx += stride) {
        int j    = idx & 1;
        int lane = (idx >> 1) & 31;
        int ks   = (idx >> 6) % 25;
        int nt   = idx / (25 * 64);
        int d    = ks * 4 + ((lane >> 4) << 1) + j;   // K index (direction)
        int c    = nt * 16 + (lane & 15);             // N index (coeff)
        pb2[idx] = (c < NC) ? sft[c * ND + d] : 0.0f;
    }
}

// ---------------------------------------------------------------------------
// nonlin: out[b,ch,:] = sft @ relu(isft @ in[b,ch,:])   as two chained GEMMs.
// One wave per 16-row tile of the flattened [B*ChTot, 48] matrix.
// Concat handled via two row sources (ChA rows from inA, ChB rows from inB);
// all channel counts are multiples of 16 so a tile never straddles a source.
// ---------------------------------------------------------------------------
__global__ __launch_bounds__(64) void nonlin_kernel(
    const float* __restrict__ inA, int chA,
    const float* __restrict__ inB, int chB,
    float* __restrict__ out,
    const float* __restrict__ pb1, const float* __restrict__ pb2) {

    __shared__ float Y[2][16][114];                  // relu(GEMM1) staging
    const int lane = threadIdx.x & 31;
    const int wave = threadIdx.x >> 5;
    const int lh   = lane >> 4;                      // 0 = lanes 0-15
    const int ll   = lane & 15;

    const int chTot   = chA + chB;
    const int tile    = blockIdx.x * 2 + wave;
    const int rowBase = tile * 16;
    const int b       = rowBase / chTot;
    const int chBase  = rowBase % chTot;
    const float* src  = (chBase < chA)
        ? (inA + (size_t)(b * chA + chBase) * NCP)
        : (inB + (size_t)(b * chB + (chBase - chA)) * NCP);

    // ---- GEMM1: Y[16,112] = X[16,48] * isft^T  (K padded with zero B rows)
    v8f acc[NDT];
#pragma unroll
    for (int nt = 0; nt < NDT; ++nt) acc[nt] = v8zero();

#pragma unroll
    for (int ks = 0; ks < 12; ++ks) {
        v2f a = *(const v2f*)(src + (size_t)ll * NCP + ks * 4 + 2 * lh);
#pragma unroll
        for (int nt = 0; nt < NDT; ++nt) {
            v2f bb = *(const v2f*)(pb1 + ((size_t)(nt * 12 + ks) * 32 + lane) * 2);
            acc[nt] = wmma4(a, bb, acc[nt]);
        }
    }

    // ---- relu, C-layout -> LDS (row = M, col = N)
#pragma unroll
    for (int nt = 0; nt < NDT; ++nt)
#pragma unroll
        for (int r = 0; r < 8; ++r) {
            float v = acc[nt][r];
            Y[wave][r + 8 * lh][nt * 16 + ll] = v > 0.0f ? v : 0.0f;
        }
    // same-wave LDS RAW: DS ops are issued and complete in order per wave.

    // ---- GEMM2: Z[16,48] = Y[16,100] * sft^T   (K = 100 exact)
    v8f acc2[3];
#pragma unroll
    for (int nt = 0; nt < 3; ++nt) acc2[nt] = v8zero();

#pragma unroll
    for (int ks = 0; ks < 25; ++ks) {
        v2f a = *(const v2f*)(&Y[wave][ll][ks * 4 + 2 * lh]);
#pragma unroll
        for (int nt = 0; nt < 3; ++nt) {
            v2f bb = *(const v2f*)(pb2 + ((size_t)(nt * 25 + ks) * 32 + lane) * 2);
            acc2[nt] = wmma4(a, bb, acc2[nt]);
        }
    }

    // ---- store (cols 45..47 are exact zeros via zero-padded sft rows)
    float* dst = out + (size_t)rowBase * NCP;
#pragma unroll
    for (int nt = 0; nt < 3; ++nt)
#pragma unroll
        for (int r = 0; r < 8; ++r)
            dst[(size_t)(r + 8 * lh) * NCP + nt * 16 + ll] = acc2[nt][r];
}

// ---------------------------------------------------------------------------
// sconv: out[b,o,c] = lfac(c) * sum_i w[o,i,ls[c]/2] * x[b,i,c]
// One wave per (batch-tile of 16, coefficient c). A tile held in registers,
// swept over all O-tiles so the input is read exactly once.
// blockIdx.y covers 48 c's; c>=45 zero-fills the padded columns.
// ---------------------------------------------------------------------------
template <int I, int O, bool LAST>
__global__ __launch_bounds__(32) void sconv_kernel(
    const float* __restrict__ x, int inStride,
    const float* __restrict__ w, const int* __restrict__ ls,
    float* __restrict__ out) {

    constexpr int KS = I / 4;
    constexpr int OT = (O + 15) / 16;
    const int c    = blockIdx.y;
    const int bt   = blockIdx.x;
    const int lane = threadIdx.x;
    const int lh   = lane >> 4;
    const int ll   = lane & 15;

    if (!LAST && c >= NC) {                 // zero padded coefficient columns
#pragma unroll
        for (int ot = 0; ot < OT; ++ot)
#pragma unroll
            for (int r = 0; r < 8; ++r) {
                int m = r + 8 * lh;
                out[(((size_t)(bt * 16 + m)) * O + ot * 16 + ll) * NCP + c] = 0.0f;
            }
        return;
    }

    const int   l    = ls[c];
    const float lfac = sqrtf(PI_F / (float)(2 * l + 1));
    const int   s    = l >> 1;

    // ---- A tile in registers: A[m=b, k=i]
    v2f a[KS];
    const float* xb = x + (size_t)(bt * 16 + ll) * I * inStride + c;
#pragma unroll
    for (int ks = 0; ks < KS; ++ks) {
        int i0 = ks * 4 + 2 * lh;
        v2f av;
        av[0] = xb[(size_t)i0 * inStride];
        av[1] = xb[(size_t)(i0 + 1) * inStride];
        a[ks] = av;
    }

#pragma unroll
    for (int ot = 0; ot < OT; ++ot) {
        const int o  = ot * 16 + ll;
        const int oc = (o < O) ? o : 0;     // clamp (only matters for O==1)
        const float* wb = w + (size_t)oc * I * 5 + s;

        v8f acc = v8zero();
#pragma unroll
        for (int ks = 0; ks < KS; ++ks) {
            int i0 = ks * 4 + 2 * lh;
            v2f bb;
            bb[0] = wb[(size_t)i0 * 5];
            bb[1] = wb[(size_t)(i0 + 1) * 5];
            acc = wmma4(a[ks], bb, acc);
        }

        if (!LAST) {
            float* dst = out + ((size_t)(bt * 16) * O + ot * 16) * NCP + c;
#pragma unroll
            for (int r = 0; r < 8; ++r) {
                int m = r + 8 * lh;
                dst[((size_t)m * O + ll) * NCP] = acc[r] * lfac;
            }
        } else {
            if (ll == 0) {                  // only o == 0 is real output
                float* dst = out + (size_t)(bt * 16) * NC + c;
#pragma unroll
                for (int r = 0; r < 8; ++r)
                    dst[(size_t)(r + 8 * lh) * NC] = acc[r] * lfac;
            }
        }
    }
}

// ---------------------------------------------------------------------------
// Workspace layout (floats)
// ---------------------------------------------------------------------------
static constexpr size_t OFF_PB1 = 0;                         // 5376
static constexpr size_t OFF_PB2 = 5376;                      // 4800
static constexpr size_t OFF_T   = 16384;                     // scratch (sconv out)
static constexpr size_t SZ_T    = (size_t)BATCH * 128 * NCP;
static constexpr size_t OFF_E1  = OFF_T + SZ_T;
static constexpr size_t SZ_E1   = (size_t)BATCH * 16 * NCP;
static constexpr size_t OFF_E2  = OFF_E1 + SZ_E1;
static constexpr size_t SZ_E2   = (size_t)BATCH * 32 * NCP;
static constexpr size_t OFF_E3  = OFF_E2 + SZ_E2;
static constexpr size_t SZ_E3   = (size_t)BATCH * 64 * NCP;
static constexpr size_t OFF_E4  = OFF_E3 + SZ_E3;
static constexpr size_t SZ_E4   = (size_t)BATCH * 128 * NCP;
static constexpr size_t WS_FLOATS = OFF_E4 + SZ_E4;          // ~1.08 GiB

static inline void launch_nonlin(const float* inA, int chA,
                                 const float* inB, int chB,
                                 float* out, const float* pb1, const float* pb2,
                                 hipStream_t stream) {
    int tiles = BATCH * (chA + chB) / 16;    // always even
    nonlin_kernel<<<tiles / 2, 64, 0, stream>>>(inA, chA, inB, chB, out, pb1, pb2);
}

extern "C" void kernel_launch(void* const* d_in, const int* in_sizes, int n_in,
                              void* d_out, int out_size, void* d_ws, size_t ws_size,
                              hipStream_t stream) {
    (void)in_sizes; (void)n_in; (void)out_size;
    if (ws_size < WS_FLOATS * sizeof(float)) return;  // deterministic no-op guard

    const float* x    = (const float*)d_in[0];
    const float* sft  = (const float*)d_in[1];
    const float* isft = (const float*)d_in[2];
    const float* w1   = (const float*)d_in[3];
    const float* w2   = (const float*)d_in[4];
    const float* w3   = (const float*)d_in[5];
    const float* w4   = (const float*)d_in[6];
    const float* w5   = (const float*)d_in[7];
    const float* w6   = (const float*)d_in[8];
    const float* w7   = (const float*)d_in[9];
    const float* w8   = (const float*)d_in[10];
    const float* w9   = (const float*)d_in[11];
    const int*   ls   = (const int*)d_in[12];

    float* ws  = (float*)d_ws;
    float* pb1 = ws + OFF_PB1;
    float* pb2 = ws + OFF_PB2;
    float* t   = ws + OFF_T;
    float* e1  = ws + OFF_E1;
    float* e2  = ws + OFF_E2;
    float* e3  = ws + OFF_E3;
    float* e4  = ws + OFF_E4;

    prep_pack<<<40, 256, 0, stream>>>(sft, isft, pb1, pb2);

    const dim3 gP(BATCH / 16, NCP);     // padded c grid
    const dim3 gL(BATCH / 16, NC);      // last layer: real c only

    // -------- encoder --------
    sconv_kernel<4,   16, false><<<gP, 32, 0, stream>>>(x,  45, w1, ls, t);
    launch_nonlin(t, 16, nullptr, 0, e1, pb1, pb2, stream);
    sconv_kernel<16,  32, false><<<gP, 32, 0, stream>>>(e1, NCP, w2, ls, t);
    launch_nonlin(t, 32, nullptr, 0, e2, pb1, pb2, stream);
    sconv_kernel<32,  64, false><<<gP, 32, 0, stream>>>(e2, NCP, w3, ls, t);
    launch_nonlin(t, 64, nullptr, 0, e3, pb1, pb2, stream);
    sconv_kernel<64, 128, false><<<gP, 32, 0, stream>>>(e3, NCP, w4, ls, t);
    launch_nonlin(t, 128, nullptr, 0, e4, pb1, pb2, stream);

    // -------- decoder (skip concats; outputs reuse dead buffers) --------
    sconv_kernel<128, 64, false><<<gP, 32, 0, stream>>>(e4, NCP, w5, ls, t);
    launch_nonlin(t, 64, e3, 64, e4, pb1, pb2, stream);        // d1 -> e4
    sconv_kernel<128, 32, false><<<gP, 32, 0, stream>>>(e4, NCP, w6, ls, t);
    launch_nonlin(t, 32, e2, 32, e3, pb1, pb2, stream);        // d2 -> e3
    sconv_kernel<64,  16, false><<<gP, 32, 0, stream>>>(e3, NCP, w7, ls, t);
    launch_nonlin(t, 16, e1, 16, e2, pb1, pb2, stream);        // d3 -> e2
    sconv_kernel<32,  16, false><<<gP, 32, 0, stream>>>(e2, NCP, w8, ls, t);
    launch_nonlin(t, 16, nullptr, 0, e1, pb1, pb2, stream);    // d4 -> e1

    // -------- head: [B,1,45] -> d_out [B,45] --------
    sconv_kernel<16, 1, true><<<gL, 32, 0, stream>>>(e1, NCP, w9, ls, (float*)d_out);
}